// BasicModelThetaPerStep_49074296324488
// MI455X (gfx1250) — compile-verified
//
#include <hip/hip_runtime.h>

typedef __attribute__((ext_vector_type(2))) float v2f;
typedef __attribute__((ext_vector_type(8))) float v8f;

constexpr int BB  = 256;
constexpr int TT  = 512;
constexpr int DIN = 65;     // COV + 1
constexpr int HH  = 128;
constexpr int H3  = 384;
constexpr int CV  = 64;
constexpr int ND  = 3;

constexpr size_t OUT_HS_OFF  = (size_t)BB * TT * ND;                    // pred first
constexpr size_t OUT_COV_OFF = OUT_HS_OFF + (size_t)BB * TT * HH;      // then hs, then cov

// D = A(16x4 f32) * B(4x16 f32) + C(16x16 f32)
__device__ __forceinline__ v8f wmma4(v2f a, v2f b, v8f c) {
    return __builtin_amdgcn_wmma_f32_16x16x4_f32(false, a, false, b, (short)0, c,
                                                 false, false);
}

__device__ __forceinline__ float sig_(float v)  { return 1.0f / (1.0f + __expf(-v)); }
__device__ __forceinline__ float tanh_(float v) { return 1.0f - 2.0f / (__expf(2.0f * v) + 1.0f); }

// ---------------------------------------------------------------------------
// Kernel 1: GRU scan. 16 blocks (one per 16-row batch tile), 256 threads.
// Wave w owns gate tiles {w, 8+w, 16+w}  ->  h columns [16w, 16w+16).
// W_ih and W_hh live entirely in LDS (CDNA5: 320KB/WGP).
// ---------------------------------------------------------------------------
__global__ __launch_bounds__(256, 1)
void gru_scan_kernel(const float* __restrict__ x,
                     const int*   __restrict__ lengths,
                     const float* __restrict__ h0,
                     const float* __restrict__ W_ih,
                     const float* __restrict__ W_hh,
                     const float* __restrict__ b_ih,
                     const float* __restrict__ b_hh,
                     float*       __restrict__ out_hs) {
    __shared__ float lds_whh[H3 * HH];    // 192 KB : W_hh row-major [384][128]
    __shared__ float lds_wih[H3 * DIN];   // 97.5KB : W_ih row-major [384][65]
    __shared__ float lds_h[16 * HH];      // 8 KB   : current hidden tile [16][128]

    const int tid  = threadIdx.x;
    const int w    = tid >> 5;        // wave 0..7
    const int lane = tid & 31;
    const int nl   = lane & 15;       // N index of B/C frags; M index of A frags
    const int hi   = lane >> 4;       // half-wave
    const int koff = hi * 2;          // K sub-offset inside a K=4 step
    const int b0   = blockIdx.x * 16; // batch tile base

    for (int i = tid; i < H3 * HH;  i += 256) lds_whh[i] = W_hh[i];
    for (int i = tid; i < H3 * DIN; i += 256) lds_wih[i] = W_ih[i];
    for (int i = tid; i < 16 * HH;  i += 256) {
        int r = i >> 7, c = i & (HH - 1);
        lds_h[i] = h0[(size_t)(b0 + r) * HH + c];
    }
    __syncthreads();

    const int cw      = w * 16;               // owned h-column base
    const int myLenA  = lengths[b0 + nl];     // mask for my A rows (m = nl)
    int len8[8];
#pragma unroll
    for (int e = 0; e < 8; ++e) len8[e] = lengths[b0 + e + hi * 8];

    const float bir = b_ih[cw + nl],            bhr = b_hh[cw + nl];
    const float biz = b_ih[HH + cw + nl],       bhz = b_hh[HH + cw + nl];
    const float bin = b_ih[2 * HH + cw + nl],   bhn = b_hh[2 * HH + cw + nl];

    // previous h for owned columns, kept in registers in C-layout
    v8f hreg;
#pragma unroll
    for (int e = 0; e < 8; ++e) hreg[e] = lds_h[(e + hi * 8) * HH + cw + nl];

    for (int t = 0; t < TT; ++t) {
        // ---------------- gi = x_t @ W_ih^T + b_ih (x masked by length) ------
        v8f air, aiz, ain;
#pragma unroll
        for (int e = 0; e < 8; ++e) { air[e] = bir; aiz[e] = biz; ain[e] = bin; }

        const bool rv = (t < myLenA);
        const float* xrow = x + ((size_t)(b0 + nl) * TT + t) * DIN;

        for (int kk = 0; kk < 17; ++kk) {
            const int k = kk * 4 + koff;
            v2f a; a.x = 0.0f; a.y = 0.0f;
            if (rv) {
                if (k < DIN)     a.x = xrow[k];
                if (k + 1 < DIN) a.y = xrow[k + 1];
            }
#pragma unroll
            for (int g = 0; g < 3; ++g) {
                const int nb = g * HH + cw;
                v2f bf;
                const float* wr = &lds_wih[(nb + nl) * DIN + k];
                bf.x = (k < DIN)     ? wr[0] : 0.0f;
                bf.y = (k + 1 < DIN) ? wr[1] : 0.0f;
                v8f& acc = (g == 0) ? air : (g == 1) ? aiz : ain;
                acc = wmma4(a, bf, acc);
            }
        }

        // ---------------- gh = h @ W_hh^T + b_hh ----------------------------
        v8f ahr, ahz, ahn;
#pragma unroll
        for (int e = 0; e < 8; ++e) { ahr[e] = bhr; ahz[e] = bhz; ahn[e] = bhn; }

        const float* hrow = &lds_h[nl * HH];
        for (int kk = 0; kk < 32; ++kk) {
            const int k = kk * 4 + koff;
            v2f a = *(const v2f*)(hrow + k);            // 8B aligned
#pragma unroll
            for (int g = 0; g < 3; ++g) {
                const int nb = g * HH + cw;
                v2f bf = *(const v2f*)(&lds_whh[(nb + nl) * HH + k]);
                v8f& acc = (g == 0) ? ahr : (g == 1) ? ahz : ahn;
                acc = wmma4(a, bf, acc);
            }
        }

        // ---------------- gating (lane-local: columns owned by this wave) ---
        v8f hnew;
#pragma unroll
        for (int e = 0; e < 8; ++e) {
            float r    = sig_(air[e] + ahr[e]);
            float z    = sig_(aiz[e] + ahz[e]);
            float n    = tanh_(ain[e] + r * ahn[e]);
            hnew[e]    = (1.0f - z) * n + z * hreg[e];
        }
        hreg = hnew;

        __syncthreads();                 // all waves done reading lds_h
#pragma unroll
        for (int e = 0; e < 8; ++e)
            lds_h[(e + hi * 8) * HH + cw + nl] = hnew[e];
        __syncthreads();                 // lds_h updated for next step

        // ---------------- store hs (masked past length) ---------------------
#pragma unroll
        for (int e = 0; e < 8; ++e) {
            const int m = e + hi * 8;
            float v = (t < len8[e]) ? hnew[e] : 0.0f;
            out_hs[((size_t)(b0 + m) * TT + t) * HH + cw + nl] = v;
        }
    }
}

// ---------------------------------------------------------------------------
// Kernel 2: output heads. One wave per 16-row (b, t0..t0+15) tile.
//   pred = exp(-(hs @ W_p^T + b_p))           [B,T,3]   (all t)
//   cov  = mask * ([hs ; x_time] @ W_c^T + b_c) [B,T-1,65] (col 64 = 0)
// ---------------------------------------------------------------------------
__global__ __launch_bounds__(256)
void heads_kernel(const float* __restrict__ x,
                  const int*   __restrict__ lengths,
                  const float* __restrict__ hs,      // masked hs (from kernel 1)
                  const float* __restrict__ W_p,
                  const float* __restrict__ b_p,
                  const float* __restrict__ W_c,
                  const float* __restrict__ b_c,
                  float*       __restrict__ out) {
    const int tid  = threadIdx.x;
    const int w    = tid >> 5;
    const int lane = tid & 31;
    const int nl   = lane & 15;
    const int hi   = lane >> 4;
    const int koff = hi * 2;

    const int tile = blockIdx.x * 8 + w;    // 0 .. B*T/16-1
    const int r0   = tile * 16;             // global row base = b*T + t0
    const int b    = r0 >> 9;               // / T
    const int t0   = r0 & (TT - 1);
    const int L    = lengths[b];

    v8f accc[4];
#pragma unroll
    for (int g = 0; g < 4; ++g) {
        const float bc = b_c[g * 16 + nl];
#pragma unroll
        for (int e = 0; e < 8; ++e) accc[g][e] = bc;
    }
    v8f accp;
    {
        const float bp = (nl < ND) ? b_p[nl] : 0.0f;
#pragma unroll
        for (int e = 0; e < 8; ++e) accp[e] = bp;
    }

    // A row m = nl  ->  time step t0+nl
    const float* hrow  = hs + ((size_t)b * TT + t0 + nl) * HH;
    const float  xfeat = ((t0 + nl) < L) ? x[((size_t)b * TT + t0 + nl) * DIN] : 0.0f;

    for (int kk = 0; kk < 33; ++kk) {
        const int k = kk * 4 + koff;
        v2f a;
        a.x = (k < HH)     ? hrow[k]     : ((k == HH)     ? xfeat : 0.0f);
        a.y = (k + 1 < HH) ? hrow[k + 1] : ((k + 1 == HH) ? xfeat : 0.0f);

        if (kk < 32) {   // pred head: K = 128 only
            v2f bp;
            if (nl < ND) bp = *(const v2f*)(W_p + nl * HH + k);
            else         { bp.x = 0.0f; bp.y = 0.0f; }
            accp = wmma4(a, bp, accp);
        }
#pragma unroll
        for (int g = 0; g < 4; ++g) {    // cov head: K = 129 (+3 zero pad)
            const float* wr = W_c + (size_t)(g * 16 + nl) * (HH + 1) + k;
            v2f bc;
            bc.x = (k     <= HH) ? wr[0] : 0.0f;
            bc.y = (k + 1 <= HH) ? wr[1] : 0.0f;
            accc[g] = wmma4(a, bc, accc[g]);
        }
    }

    // ---- store pred_params: exp(-v), all t --------------------------------
#pragma unroll
    for (int e = 0; e < 8; ++e) {
        const int m = e + hi * 8;
        if (nl < ND)
            out[(size_t)(r0 + m) * ND + nl] = __expf(-accp[e]);
    }

    // ---- store cov: rows t in [0, T-1), zero where t >= len-1 -------------
#pragma unroll
    for (int e = 0; e < 8; ++e) {
        const int m = e + hi * 8;
        const int t = t0 + m;
        if (t < TT - 1) {
            const size_t rbase = OUT_COV_OFF + ((size_t)b * (TT - 1) + t) * (CV + 1);
            const bool mm = (t < L - 1);
#pragma unroll
            for (int g = 0; g < 4; ++g)
                out[rbase + g * 16 + nl] = mm ? accc[g][e] : 0.0f;
        }
    }
    // extra column (index 64) stays zero
    if (lane < 16) {
        const int t = t0 + lane;
        if (t < TT - 1)
            out[OUT_COV_OFF + ((size_t)b * (TT - 1) + t) * (CV + 1) + CV] = 0.0f;
    }
}

extern "C" void kernel_launch(void* const* d_in, const int* in_sizes, int n_in,
                              void* d_out, int out_size, void* d_ws, size_t ws_size,
                              hipStream_t stream) {
    const float* x      = (const float*)d_in[0];
    const int*   len    = (const int*)  d_in[1];
    const float* h0     = (const float*)d_in[2];
    const float* W_ih   = (const float*)d_in[3];
    const float* W_hh   = (const float*)d_in[4];
    const float* b_ih   = (const float*)d_in[5];
    const float* b_hh   = (const float*)d_in[6];
    const float* W_p    = (const float*)d_in[7];
    const float* b_p    = (const float*)d_in[8];
    const float* W_c    = (const float*)d_in[9];
    const float* b_c    = (const float*)d_in[10];
    float* out = (float*)d_out;

    gru_scan_kernel<<<BB / 16, 256, 0, stream>>>(x, len, h0, W_ih, W_hh, b_ih, b_hh,
                                                 out + OUT_HS_OFF);
    heads_kernel<<<(BB * TT / 16) / 8, 256, 0, stream>>>(x, len, out + OUT_HS_OFF,
                                                         W_p, b_p, W_c, b_c, out);
}